// DeepIRTModel_9801115369543
// MI455X (gfx1250) — compile-verified
//
#include <hip/hip_runtime.h>
#include <hip/hip_bf16.h>

// ---------------- problem constants ----------------
constexpr int NQ = 50000;
constexpr int Mm = 50;    // memory slots
constexpr int DK = 50;
constexpr int DV = 200;
constexpr int DS = 50;
constexpr int Bb = 256;
constexpr int Tt = 200;
constexpr int R  = Bb * Tt;     // 51200 flattened (b,t) rows
constexpr float SCALE = 3.0f;

typedef __attribute__((ext_vector_type(2))) float v2f;
typedef __attribute__((ext_vector_type(8))) float v8f;

__device__ __forceinline__ v8f wmma_f32(v2f a, v2f b, v8f c) {
    // V_WMMA_F32_16X16X4_F32 : D(16x16,f32) = A(16x4,f32) * B(4x16,f32) + C
    return __builtin_amdgcn_wmma_f32_16x16x4_f32(
        /*neg_a=*/false, a, /*neg_b=*/false, b,
        /*c_mod=*/(short)0, c, /*reuse_a=*/false, /*reuse_b=*/false);
}

__device__ __forceinline__ float sigmoidf_(float x) {
    return 1.0f / (1.0f + __expf(-x));
}

// ---------------------------------------------------------------------------
// Phase 1a: er = sigmoid(qae@We + be), ad = tanh(qae@Wad + bad)
// One wave per (16-row, 16-col) tile; A fragment gathered from qa_tab.
// A layout (16x4 f32): lane L -> row L%16, k-pair base 2*(L/16).
// C layout (16x16 f32): vgpr v, lane L -> row v + 8*(L/16), col L%16.
// ---------------------------------------------------------------------------
__global__ __launch_bounds__(32) void k_erad(
    const int* __restrict__ qa_data, const float* __restrict__ qa_tab,
    const float* __restrict__ We,  const float* __restrict__ be,
    const float* __restrict__ Wad, const float* __restrict__ bad,
    float* __restrict__ er, float* __restrict__ ad)
{
    const int rt   = blockIdx.x;          // 0..R/16-1
    const int ct   = blockIdx.y;          // 0..12 (13 col tiles over DV=200)
    const int lane = threadIdx.x;         // 0..31
    const int l16  = lane & 15;
    const int half = lane >> 4;           // 0 or 1
    const int row  = rt * 16 + l16;
    const int col  = ct * 16 + l16;
    const bool colOK = (col < DV);

    const long aBase = (long)qa_data[row] * DV;   // gathered embedding row

    v8f accE = {};
    v8f accA = {};
    for (int k0 = 0; k0 < DV; k0 += 4) {
        const int ka = k0 + 2 * half;
        v2f a;
        a.x = qa_tab[aBase + ka];
        a.y = qa_tab[aBase + ka + 1];
        v2f bE = {0.0f, 0.0f}, bA = {0.0f, 0.0f};
        if (colOK) {
            bE.x = We[(long)ka * DV + col];
            bE.y = We[(long)(ka + 1) * DV + col];
            bA.x = Wad[(long)ka * DV + col];
            bA.y = Wad[(long)(ka + 1) * DV + col];
        }
        accE = wmma_f32(a, bE, accE);
        accA = wmma_f32(a, bA, accA);
    }

    if (colOK) {
        const float biasE = be[col];
        const float biasA = bad[col];
        #pragma unroll
        for (int v = 0; v < 8; ++v) {
            const int rr = rt * 16 + v + 8 * half;
            er[(long)rr * DV + col] = sigmoidf_(accE[v] + biasE);
            ad[(long)rr * DV + col] = tanhf(accA[v] + biasA);
        }
    }
}

// ---------------------------------------------------------------------------
// Phase 1b: logits = qe@Mk^T  (raw, softmax later)   [R x 50]
//           hd     = tanh(qe@Wd1 + bd1)              [R x 50]
// One wave per 16-row tile; all 4 col-tiles (cols 0..63, guard <50) per wave.
// K = 50 padded to 52 with zero fill.
// ---------------------------------------------------------------------------
__global__ __launch_bounds__(32) void k_logits(
    const int* __restrict__ q_data, const float* __restrict__ q_tab,
    const float* __restrict__ Mk, const float* __restrict__ Wd1,
    const float* __restrict__ bd1,
    float* __restrict__ logits, float* __restrict__ hd)
{
    const int rt   = blockIdx.x;
    const int lane = threadIdx.x;
    const int l16  = lane & 15;
    const int half = lane >> 4;
    const int row  = rt * 16 + l16;

    const long aBase = (long)q_data[row] * DK;

    v8f accL[4] = {};
    v8f accH[4] = {};
    for (int k0 = 0; k0 < 52; k0 += 4) {
        const int ka = k0 + 2 * half;
        v2f a;
        a.x = (ka     < DK) ? q_tab[aBase + ka]     : 0.0f;
        a.y = (ka + 1 < DK) ? q_tab[aBase + ka + 1] : 0.0f;
        #pragma unroll
        for (int ct = 0; ct < 4; ++ct) {
            const int col = ct * 16 + l16;
            v2f bL = {0.0f, 0.0f}, bH = {0.0f, 0.0f};
            if (col < Mm) {  // Mm == DS == 50
                if (ka < DK) {
                    bL.x = Mk[(long)col * DK + ka];        // Mk^T: B[k][n] = Mk[n][k]
                    bH.x = Wd1[(long)ka * DS + col];
                }
                if (ka + 1 < DK) {
                    bL.y = Mk[(long)col * DK + ka + 1];
                    bH.y = Wd1[(long)(ka + 1) * DS + col];
                }
            }
            accL[ct] = wmma_f32(a, bL, accL[ct]);
            accH[ct] = wmma_f32(a, bH, accH[ct]);
        }
    }

    #pragma unroll
    for (int ct = 0; ct < 4; ++ct) {
        const int col = ct * 16 + l16;
        if (col < Mm) {
            const float b1 = bd1[col];
            #pragma unroll
            for (int v = 0; v < 8; ++v) {
                const int rr = rt * 16 + v + 8 * half;
                logits[(long)rr * Mm + col] = accL[ct][v];
                hd[(long)rr * DS + col]     = tanhf(accH[ct][v] + b1);
            }
        }
    }
}

// ---------------------------------------------------------------------------
// Phase 1c: in-place softmax over M on w-buffer, and beta = hd@Wd2 + bd2.
// One thread per row.
// ---------------------------------------------------------------------------
__global__ __launch_bounds__(256) void k_softmax_beta(
    const float* __restrict__ Wd2, const float* __restrict__ bd2,
    float* __restrict__ w, const float* __restrict__ hd,
    float* __restrict__ beta)
{
    const int r = blockIdx.x * blockDim.x + threadIdx.x;
    if (r >= R) return;
    float lg[Mm];
    float mx = -1e30f;
    #pragma unroll 1
    for (int n = 0; n < Mm; ++n) { lg[n] = w[(long)r * Mm + n]; mx = fmaxf(mx, lg[n]); }
    float s = 0.0f;
    #pragma unroll 1
    for (int n = 0; n < Mm; ++n) { lg[n] = __expf(lg[n] - mx); s += lg[n]; }
    const float inv = 1.0f / s;
    #pragma unroll 1
    for (int n = 0; n < Mm; ++n) w[(long)r * Mm + n] = lg[n] * inv;

    float acc = bd2[0];
    #pragma unroll 1
    for (int sI = 0; sI < DS; ++sI) acc += hd[(long)r * DS + sI] * Wd2[sI];
    beta[r] = acc;
}

// ---------------------------------------------------------------------------
// Phase 2: sequential scan, one workgroup (256 thr = 8 waves) per batch elem.
// Mv state (50x200 f32 = 40KB) lives in LDS for all 200 steps.
// ---------------------------------------------------------------------------
__global__ __launch_bounds__(256) void k_scan(
    const int* __restrict__ q_data, const float* __restrict__ q_tab,
    const float* __restrict__ Mv0,
    const float* __restrict__ Ws,  const float* __restrict__ bs,
    const float* __restrict__ Wa1, const float* __restrict__ ba1,
    const float* __restrict__ Wa2, const float* __restrict__ ba2,
    const float* __restrict__ er, const float* __restrict__ ad,
    const float* __restrict__ w,  const float* __restrict__ beta,
    float* __restrict__ out)
{
    __shared__ float Mv[Mm * DV];        // 40 KB batch-private state
    __shared__ float wt_s[Mm];
    __shared__ float qe_s[DK];
    __shared__ float er_s[DV];
    __shared__ float ad_s[DV];
    __shared__ float read_s[DV];
    __shared__ float summ_s[DS];
    __shared__ float h_s[DS];

    const int b   = blockIdx.x;
    const int tid = threadIdx.x;

    for (int e = tid; e < Mm * DV; e += 256) Mv[e] = Mv0[e];
    __syncthreads();

    for (int t = 0; t < Tt; ++t) {
        const long r = (long)b * Tt + t;

        if (tid < Mm) wt_s[tid] = w[r * Mm + tid];
        if (tid < DK) {
            const int idx = q_data[r];
            qe_s[tid] = q_tab[(long)idx * DK + tid];
        }
        if (tid < DV) {
            er_s[tid] = er[r * DV + tid];
            ad_s[tid] = ad[r * DV + tid];
        }
        __syncthreads();

        // read[d] = sum_n w[n] * Mv[n][d]   (consecutive d across threads:
        // conflict-free LDS banks)
        if (tid < DV) {
            float acc = 0.0f;
            #pragma unroll 5
            for (int n = 0; n < Mm; ++n) acc += wt_s[n] * Mv[n * DV + tid];
            read_s[tid] = acc;
        }
        __syncthreads();

        // Mv <- Mv*(1 - w⊗er) + w⊗ad   (uses old Mv; safe after read barrier)
        for (int e = tid; e < Mm * DV; e += 256) {
            const int n = e / DV, d = e - n * DV;
            const float wv = wt_s[n];
            Mv[e] = Mv[e] * (1.0f - wv * er_s[d]) + wv * ad_s[d];
        }
        // summ = tanh([read, qe] @ Ws + bs)
        if (tid < DS) {
            float acc = bs[tid];
            #pragma unroll 4
            for (int j = 0; j < DV; ++j) acc += read_s[j] * Ws[j * DS + tid];
            #pragma unroll 5
            for (int j = 0; j < DK; ++j) acc += qe_s[j] * Ws[(DV + j) * DS + tid];
            summ_s[tid] = tanhf(acc);
        }
        __syncthreads();

        if (tid < DS) {
            float acc = ba1[tid];
            #pragma unroll 5
            for (int s = 0; s < DS; ++s) acc += summ_s[s] * Wa1[s * DS + tid];
            h_s[tid] = tanhf(acc);
        }
        __syncthreads();

        if (tid == 0) {
            float acc = ba2[0];
            #pragma unroll 5
            for (int s = 0; s < DS; ++s) acc += h_s[s] * Wa2[s];
            out[r] = sigmoidf_(SCALE * acc - beta[r]);
        }
        __syncthreads();
    }
}

// ---------------------------------------------------------------------------
extern "C" void kernel_launch(void* const* d_in, const int* in_sizes, int n_in,
                              void* d_out, int out_size, void* d_ws, size_t ws_size,
                              hipStream_t stream)
{
    const int*   q_data  = (const int*)  d_in[0];
    const int*   qa_data = (const int*)  d_in[1];
    const float* q_tab   = (const float*)d_in[2];
    const float* qa_tab  = (const float*)d_in[3];
    const float* Mk      = (const float*)d_in[4];
    const float* Mv0     = (const float*)d_in[5];
    const float* Ws      = (const float*)d_in[6];
    const float* bs      = (const float*)d_in[7];
    const float* Wa1     = (const float*)d_in[8];
    const float* ba1     = (const float*)d_in[9];
    const float* Wa2     = (const float*)d_in[10];
    const float* ba2     = (const float*)d_in[11];
    const float* Wd1     = (const float*)d_in[12];
    const float* bd1     = (const float*)d_in[13];
    const float* Wd2     = (const float*)d_in[14];
    const float* bd2     = (const float*)d_in[15];
    const float* We      = (const float*)d_in[16];
    const float* be      = (const float*)d_in[17];
    const float* Wad     = (const float*)d_in[18];
    const float* bad     = (const float*)d_in[19];

    // workspace layout (floats): er | ad | w(logits->softmax) | hd | beta
    float* ws_f  = (float*)d_ws;
    float* er    = ws_f;
    float* ad_   = er   + (size_t)R * DV;
    float* wbuf  = ad_  + (size_t)R * DV;
    float* hd    = wbuf + (size_t)R * Mm;
    float* beta  = hd   + (size_t)R * DS;
    // total = 2*R*DV + R*Mm + R*DS + R = 25,651,200 floats (~103 MB)

    k_erad<<<dim3(R / 16, (DV + 15) / 16), 32, 0, stream>>>(
        qa_data, qa_tab, We, be, Wad, bad, er, ad_);

    k_logits<<<R / 16, 32, 0, stream>>>(
        q_data, q_tab, Mk, Wd1, bd1, wbuf, hd);

    k_softmax_beta<<<(R + 255) / 256, 256, 0, stream>>>(
        Wd2, bd2, wbuf, hd, beta);

    k_scan<<<Bb, 256, 0, stream>>>(
        q_data, q_tab, Mv0, Ws, bs, Wa1, ba1, Wa2, ba2,
        er, ad_, wbuf, beta, (float*)d_out);
}